// MDlayer_34059090657518
// MI455X (gfx1250) — compile-verified
//
#include <hip/hip_runtime.h>
#include <stdint.h>

// Problem dims (fixed by the reference): x[256,1024], Wmin/Wmax[512,1024] -> out[256,512]
#define B_DIM 256
#define U_DIM 512
#define D_DIM 1024

// Tiling: 64x64 output tile, 256 threads (8 wave32), 4x4 micro-tile per thread.
#define BM 64
#define BN 64
#define BK 32
#define KSPLIT 8                    // split-K for occupancy: 256 blocks total
#define KCHUNK (D_DIM / KSPLIT)     // 128 k per block
#define NCHUNK (KCHUNK / BK)        // 4 double-buffered chunks
#define RS (BK + 1)                 // LDS row stride (TDM pad: +1 dword) -> conflict-free
#define TILE (BM * RS)              // 2112 floats per tile
#define TILE_BYTES (TILE * 4)       // 8448 bytes

typedef unsigned int tdm_u32x4 __attribute__((ext_vector_type(4)));
typedef int          tdm_i32x8 __attribute__((ext_vector_type(8)));
typedef int          tdm_i32x4 __attribute__((ext_vector_type(4)));

// Loop-invariant part of the TDM D# for our f32 2D tiles:
// group1 dword0: workgroup_mask=0 | data_size=2 (4B) | pad_enable
//                | pad_interval code 4 (=32 dwords/row) | pad_amount code 0 (=1 dword)
__device__ __forceinline__ tdm_i32x8 tdm_make_g1(unsigned int tile_w,
                                                 unsigned int tile_h,
                                                 unsigned int row_stride_elems)
{
    tdm_i32x8 g1;
    g1[0] = (int)((2u << 16) | (1u << 20) | (4u << 22));
    g1[1] = (int)((tile_w & 0xFFFFu) << 16);                   // abar=0 | tensor_dim0[15:0]
    g1[2] = (int)((tile_w >> 16) | ((tile_h & 0xFFFFu) << 16));// tensor_dim0[31:16] | tensor_dim1[15:0]
    g1[3] = (int)((tile_h >> 16) | (tile_w << 16));            // tensor_dim1[31:16] | tile_dim0
    g1[4] = (int)(tile_h & 0xFFFFu);                           // tile_dim1 | tile_dim2=0 (2D)
    g1[5] = (int)row_stride_elems;                             // tensor_dim0_stride[31:0]
    g1[6] = 0;                                                 // stride[47:32] | dim1_stride[15:0]=0
    g1[7] = 0;
    return g1;
}

// Issue TENSOR_LOAD_TO_LDS for one tile: patch lds_addr / global_addr into group0.
__device__ __forceinline__ void tdm_issue(unsigned int lds_byte_addr,
                                          const void* gptr,
                                          tdm_i32x8 g1)
{
    unsigned long long ga = (unsigned long long)(uintptr_t)gptr;

    tdm_u32x4 g0;
    g0.x = 1u;                                        // count=1 (valid user descriptor)
    g0.y = lds_byte_addr;                             // lds_addr (bytes)
    g0.z = (unsigned int)(ga & 0xFFFFFFFFull);        // global_addr[31:0]
    g0.w = (unsigned int)((ga >> 32) & 0x01FFFFFFull) // global_addr[56:32]
         | (2u << 30);                                // type = 2 ("image")

    tdm_i32x4 z4 = {0, 0, 0, 0};
    tdm_i32x8 z8 = {0, 0, 0, 0, 0, 0, 0, 0};
    __builtin_amdgcn_tensor_load_to_lds(g0, g1, z4, z4, z8, 0);
}

#define FMIN(a, b) __builtin_fminf((a), (b))
#define UPD(acc, xv, nv, mv) acc = FMIN(acc, FMIN((xv) - (nv), (mv) - (xv)))

// One k-step of the 4x4 micro-kernel: 12 LDS dwords in, 16 accumulator updates.
#define KBODY(K)                                                               \
    do {                                                                       \
        const float xv0 = smem[xi + 0 * RS + (K)];                             \
        const float xv1 = smem[xi + 1 * RS + (K)];                             \
        const float xv2 = smem[xi + 2 * RS + (K)];                             \
        const float xv3 = smem[xi + 3 * RS + (K)];                             \
        const float nv0 = smem[ni + 0 * RS + (K)];                             \
        const float nv1 = smem[ni + 1 * RS + (K)];                             \
        const float nv2 = smem[ni + 2 * RS + (K)];                             \
        const float nv3 = smem[ni + 3 * RS + (K)];                             \
        const float mv0 = smem[mi + 0 * RS + (K)];                             \
        const float mv1 = smem[mi + 1 * RS + (K)];                             \
        const float mv2 = smem[mi + 2 * RS + (K)];                             \
        const float mv3 = smem[mi + 3 * RS + (K)];                             \
        UPD(acc00, xv0, nv0, mv0); UPD(acc01, xv0, nv1, mv1);                  \
        UPD(acc02, xv0, nv2, mv2); UPD(acc03, xv0, nv3, mv3);                  \
        UPD(acc10, xv1, nv0, mv0); UPD(acc11, xv1, nv1, mv1);                  \
        UPD(acc12, xv1, nv2, mv2); UPD(acc13, xv1, nv3, mv3);                  \
        UPD(acc20, xv2, nv0, mv0); UPD(acc21, xv2, nv1, mv1);                  \
        UPD(acc22, xv2, nv2, mv2); UPD(acc23, xv2, nv3, mv3);                  \
        UPD(acc30, xv3, nv0, mv0); UPD(acc31, xv3, nv1, mv1);                  \
        UPD(acc32, xv3, nv2, mv2); UPD(acc33, xv3, nv3, mv3);                  \
    } while (0)

__global__ void __launch_bounds__(256)
minplus_mdlayer_kernel(const float* __restrict__ x,
                       const float* __restrict__ wmin,
                       const float* __restrict__ wmax,
                       float* __restrict__ out)
{
    __shared__ float smem[6 * TILE];   // {x, wmin, wmax} tiles, double buffered (~50 KB)

    const int tid  = threadIdx.x;
    const int tx4  = (tid & 15) * 4;   // u direction (16 threads x 4 cols)
    const int ty4  = (tid >> 4) * 4;   // b direction (16 threads x 4 rows)
    const int wave = tid >> 5;         // wave32

    const int ub = blockIdx.x * BN;            // u tile origin
    const int bb = blockIdx.y * BM;            // b tile origin
    const int k0 = blockIdx.z * KCHUNK;        // split-K origin

    const unsigned int lds_base = (unsigned int)(uintptr_t)(void*)&smem[0];

    const float* xg = x    + (size_t)bb * D_DIM + k0;
    const float* ng = wmin + (size_t)ub * D_DIM + k0;
    const float* mg = wmax + (size_t)ub * D_DIM + k0;

    // 16 accumulators as explicit scalars -> guaranteed VGPR residency.
    float acc00, acc01, acc02, acc03, acc10, acc11, acc12, acc13;
    float acc20, acc21, acc22, acc23, acc30, acc31, acc32, acc33;
    acc00 = acc01 = acc02 = acc03 = __builtin_inff();
    acc10 = acc11 = acc12 = acc13 = __builtin_inff();
    acc20 = acc21 = acc22 = acc23 = __builtin_inff();
    acc30 = acc31 = acc32 = acc33 = __builtin_inff();

    const tdm_i32x8 g1 = tdm_make_g1(BK, BM, D_DIM);   // BM == BN: same for all 3 streams

    // Prologue: wave 0 issues TDM loads for chunk 0 into buffer 0.
    if (wave == 0) {
        tdm_issue(lds_base + 0 * TILE_BYTES, xg, g1);
        tdm_issue(lds_base + 1 * TILE_BYTES, ng, g1);
        tdm_issue(lds_base + 2 * TILE_BYTES, mg, g1);
    }

#pragma unroll 1   // exactly one pipeline-stage copy
    for (int c = 0; c < NCHUNK; ++c) {
        const int buf = c & 1;

        if (wave == 0) {
            if (c + 1 < NCHUNK) {
                const int nbuf = (c + 1) & 1;
                const int ko   = (c + 1) * BK;
                tdm_issue(lds_base + (nbuf * 3 + 0) * TILE_BYTES, xg + ko, g1);
                tdm_issue(lds_base + (nbuf * 3 + 1) * TILE_BYTES, ng + ko, g1);
                tdm_issue(lds_base + (nbuf * 3 + 2) * TILE_BYTES, mg + ko, g1);
                // allow the 3 just-issued loads to stay in flight; drain chunk c
                __builtin_amdgcn_s_wait_tensorcnt(3);
            } else {
                __builtin_amdgcn_s_wait_tensorcnt(0);
            }
        }
        __syncthreads();   // chunk c data visible to all waves

        // Integer offsets into the __shared__ array keep everything addrspace(3).
        const int xi = (buf * 3 + 0) * TILE + ty4 * RS;
        const int ni = (buf * 3 + 1) * TILE + tx4 * RS;
        const int mi = (buf * 3 + 2) * TILE + tx4 * RS;

        // Manual 4-way k unroll inside an unroll-disabled loop: bounds live
        // values to 48 loaded dwords + 16 accs (no spills), and stops AMDGPU's
        // LDS-driven unroller from flattening all 32 k-steps (which spilled).
#pragma unroll 1
        for (int kk = 0; kk < BK; kk += 4) {
            KBODY(kk + 0);
            KBODY(kk + 1);
            KBODY(kk + 2);
            KBODY(kk + 3);
        }
        __syncthreads();   // everyone done reading buf before TDM reuses it
    }

    // Split-K combine: hardware f32 atomic min (global_atomic_min_num_f32).
    float* orow0 = out + (size_t)(bb + ty4 + 0) * U_DIM + ub + tx4;
    float* orow1 = out + (size_t)(bb + ty4 + 1) * U_DIM + ub + tx4;
    float* orow2 = out + (size_t)(bb + ty4 + 2) * U_DIM + ub + tx4;
    float* orow3 = out + (size_t)(bb + ty4 + 3) * U_DIM + ub + tx4;
#define AMIN(p, v) __hip_atomic_fetch_min((p), (v), __ATOMIC_RELAXED, __HIP_MEMORY_SCOPE_AGENT)
    AMIN(&orow0[0], acc00); AMIN(&orow0[1], acc01); AMIN(&orow0[2], acc02); AMIN(&orow0[3], acc03);
    AMIN(&orow1[0], acc10); AMIN(&orow1[1], acc11); AMIN(&orow1[2], acc12); AMIN(&orow1[3], acc13);
    AMIN(&orow2[0], acc20); AMIN(&orow2[1], acc21); AMIN(&orow2[2], acc22); AMIN(&orow2[3], acc23);
    AMIN(&orow3[0], acc30); AMIN(&orow3[1], acc31); AMIN(&orow3[2], acc32); AMIN(&orow3[3], acc33);
#undef AMIN
}

__global__ void init_out_kernel(float* __restrict__ out, int n)
{
    int i = blockIdx.x * blockDim.x + threadIdx.x;
    if (i < n) out[i] = __builtin_inff();
}

extern "C" void kernel_launch(void* const* d_in, const int* in_sizes, int n_in,
                              void* d_out, int out_size, void* d_ws, size_t ws_size,
                              hipStream_t stream)
{
    (void)in_sizes; (void)n_in; (void)d_ws; (void)ws_size;

    const float* x    = (const float*)d_in[0];
    const float* wmin = (const float*)d_in[1];
    const float* wmax = (const float*)d_in[2];
    float* out = (float*)d_out;

    // out must start at +inf (harness poisons it; split-K uses atomic min)
    init_out_kernel<<<(out_size + 255) / 256, 256, 0, stream>>>(out, out_size);

    dim3 grid(U_DIM / BN, B_DIM / BM, KSPLIT);   // 8 x 4 x 8 = 256 blocks
    minplus_mdlayer_kernel<<<grid, dim3(256), 0, stream>>>(x, wmin, wmax, out);
}